// NeighborhoodSimilarityGCN_85547158602130
// MI455X (gfx1250) — compile-verified
//
#include <hip/hip_runtime.h>
#include <hip/hip_bf16.h>

// ---------------------------------------------------------------------------
// 2-layer GCN for MI455X (gfx1250, wave32).
//   H = X @ W1                      -> fp32 WMMA (V_WMMA_F32_16X16X4_F32)
//   agg1 = D^-1/2 A D^-1/2 H        -> CSR-by-dst gather (H fits in 192MB L2)
//   y    = relu(agg1 + b1) @ W2     -> fused into gather kernel (per-node dot)
//   out  = D^-1/2 A D^-1/2 y + b2   -> scalar gather
// ---------------------------------------------------------------------------

#define NN 100000
#define NE 3200000
#define IND 256
#define HID 128

typedef __attribute__((ext_vector_type(2))) float v2f;
typedef __attribute__((ext_vector_type(8))) float v8f;

// ---------------- degree / CSR construction ----------------

__global__ void init_cnt_kernel(int* cnt, int n) {
    int i = blockIdx.x * blockDim.x + threadIdx.x;
    if (i < n) cnt[i] = 0;
}

__global__ void count_kernel(const int* __restrict__ dst, int* cnt, int e) {
    int i = blockIdx.x * blockDim.x + threadIdx.x;
    if (i < e) atomicAdd(&cnt[dst[i]], 1);
}

__global__ void dinv_kernel(const int* __restrict__ cnt, float* dinv, int n) {
    int i = blockIdx.x * blockDim.x + threadIdx.x;
    if (i < n) {
        float deg = (float)(cnt[i] + 1);  // +1 self loop; always > 0
        dinv[i] = rsqrtf(deg);
    }
}

// single-block exclusive scan (Hillis-Steele over 1024-chunks with carry)
__global__ void scan_kernel(const int* __restrict__ cnt, int* row_ptr, int* cursor, int n) {
    __shared__ int buf[1024];
    __shared__ int carry_s;
    int tid = threadIdx.x;
    if (tid == 0) carry_s = 0;
    __syncthreads();
    for (int base = 0; base < n; base += 1024) {
        int i = base + tid;
        int v = (i < n) ? cnt[i] : 0;
        buf[tid] = v;
        __syncthreads();
        for (int off = 1; off < 1024; off <<= 1) {
            int t = (tid >= off) ? buf[tid - off] : 0;
            __syncthreads();
            buf[tid] += t;
            __syncthreads();
        }
        int excl = buf[tid] - v + carry_s;
        if (i < n) { row_ptr[i] = excl; cursor[i] = excl; }
        int total = buf[1023] + carry_s;
        __syncthreads();
        if (tid == 0) carry_s = total;
        __syncthreads();
    }
    if (tid == 0) row_ptr[n] = carry_s;
}

__global__ void scatter_kernel(const int* __restrict__ src, const int* __restrict__ dst,
                               int* cursor, int* srcs, int e) {
    int i = blockIdx.x * blockDim.x + threadIdx.x;
    if (i < e) {
        int pos = atomicAdd(&cursor[dst[i]], 1);
        srcs[pos] = src[i];
    }
}

// ---------------- GEMM1: H[N,128] = X[N,256] @ W1[256,128] (fp32 WMMA) -----
// One block = 8 waves = one 16-row M-tile; wave w owns N-columns [16w,16w+16).
// A tile staged in LDS with row stride 260 dwords: column reads by 16 lanes
// hit banks (4*m + k) mod 64 -> conflict free (two disjoint 16-bank sets).
__global__ void __launch_bounds__(256) gemm1_kernel(const float* __restrict__ X,
                                                    const float* __restrict__ W1,
                                                    float* __restrict__ H) {
    __shared__ float lds_a[16 * 260];
    const int tid  = threadIdx.x;
    const int row0 = blockIdx.x * 16;

    // cooperative A-tile load: thread t -> row t>>4, 16 consecutive cols
    {
        int r  = tid >> 4;
        int c0 = (tid & 15) * 16;
        const float* s = X + (size_t)(row0 + r) * IND + c0;
        float*       d = lds_a + r * 260 + c0;
#pragma unroll
        for (int j = 0; j < 16; j += 4)
            *(float4*)(d + j) = *(const float4*)(s + j);
    }
    __syncthreads();

    const int lane  = tid & 31;
    const int n0    = (tid >> 5) * 16;       // wave's N offset
    const int m     = lane & 15;             // A row for this lane
    const int koff  = (lane >> 4) * 2;       // lanes 16-31 hold K=2,3
    const int nlane = lane & 15;             // B/C/D column

    v8f acc = {};
#pragma unroll 4
    for (int k0 = 0; k0 < IND; k0 += 4) {
        v2f a, b;
        a.x = lds_a[m * 260 + k0 + koff];
        a.y = lds_a[m * 260 + k0 + koff + 1];
        b.x = W1[(size_t)(k0 + koff) * HID + n0 + nlane];
        b.y = W1[(size_t)(k0 + koff + 1) * HID + n0 + nlane];
        acc = __builtin_amdgcn_wmma_f32_16x16x4_f32(
            false, a, false, b, (short)0, acc, false, false);
    }

    // C/D layout: VGPR v -> M = v + 8*(lane>=16), N = lane&15
    const int mrow = row0 + (lane >> 4) * 8;
#pragma unroll
    for (int v = 0; v < 8; ++v)
        H[(size_t)(mrow + v) * HID + n0 + nlane] = acc[v];
}

// ---------------- layer-1 aggregation fused with relu + (.)@W2 -------------
// One wave per destination node; 32 lanes x float4 = 128 features.
// Batch 32 edge (src, dinv[src]) pairs per pass, broadcast via shfl.
__global__ void __launch_bounds__(256) agg1_dot_kernel(
        const float* __restrict__ H, const float* __restrict__ dinv,
        const int* __restrict__ row_ptr, const int* __restrict__ srcs,
        const float* __restrict__ b1, const float* __restrict__ W2,
        float* __restrict__ y, int n) {
    int wave = (blockIdx.x * blockDim.x + threadIdx.x) >> 5;
    int lane = threadIdx.x & 31;
    if (wave >= n) return;

    const int start = row_ptr[wave];
    const int end   = row_ptr[wave + 1];

    float4 acc = make_float4(0.f, 0.f, 0.f, 0.f);
    for (int j0 = start; j0 < end; j0 += 32) {
        int myj = j0 + lane;
        int s_l = 0; float w_l = 0.f;
        if (myj < end) { s_l = srcs[myj]; w_l = dinv[s_l]; }
        int cnt = min(32, end - j0);
        for (int t = 0; t < cnt; ++t) {
            int   s = __shfl(s_l, t);
            float w = __shfl(w_l, t);
            float4 hs = *(const float4*)(H + (size_t)s * HID + lane * 4);
            acc.x += hs.x * w; acc.y += hs.y * w;
            acc.z += hs.z * w; acc.w += hs.w * w;
        }
    }

    const float di  = dinv[wave];
    const float dii = di * di;
    float4 hi = *(const float4*)(H + (size_t)wave * HID + lane * 4);
    float4 bb = *(const float4*)(b1 + lane * 4);
    float4 w2 = *(const float4*)(W2 + lane * 4);

    float rx = fmaxf(acc.x * di + hi.x * dii + bb.x, 0.f);
    float ry = fmaxf(acc.y * di + hi.y * dii + bb.y, 0.f);
    float rz = fmaxf(acc.z * di + hi.z * dii + bb.z, 0.f);
    float rw = fmaxf(acc.w * di + hi.w * dii + bb.w, 0.f);
    float p = rx * w2.x + ry * w2.y + rz * w2.z + rw * w2.w;

    for (int off = 16; off > 0; off >>= 1) p += __shfl_xor(p, off);
    if (lane == 0) y[wave] = p;
}

// ---------------- layer-2 aggregation (scalar per node) --------------------
__global__ void __launch_bounds__(256) agg2_kernel(
        const float* __restrict__ y, const float* __restrict__ dinv,
        const int* __restrict__ row_ptr, const int* __restrict__ srcs,
        const float* __restrict__ b2, float* __restrict__ out, int n) {
    int wave = (blockIdx.x * blockDim.x + threadIdx.x) >> 5;
    int lane = threadIdx.x & 31;
    if (wave >= n) return;

    const int start = row_ptr[wave];
    const int end   = row_ptr[wave + 1];
    float acc = 0.f;
    for (int j = start + lane; j < end; j += 32) {
        int s = srcs[j];
        acc += dinv[s] * y[s];
    }
    for (int off = 16; off > 0; off >>= 1) acc += __shfl_xor(acc, off);
    if (lane == 0) {
        float di = dinv[wave];
        out[wave] = di * acc + di * di * y[wave] + b2[0];
    }
}

// ---------------------------------------------------------------------------

extern "C" void kernel_launch(void* const* d_in, const int* in_sizes, int n_in,
                              void* d_out, int out_size, void* d_ws, size_t ws_size,
                              hipStream_t stream) {
    const float* X   = (const float*)d_in[0];
    const int*   ei  = (const int*)d_in[1];
    const float* W1  = (const float*)d_in[2];
    const float* b1  = (const float*)d_in[3];
    const float* W2  = (const float*)d_in[4];
    const float* b2  = (const float*)d_in[5];
    float* out = (float*)d_out;

    const int n = in_sizes[0] / IND;   // 100000
    const int e = in_sizes[1] / 2;     // 3200000
    const int* src = ei;
    const int* dst = ei + e;

    // workspace carve-out (16B-aligned chunks first)
    float* H       = (float*)d_ws;                 // n*HID floats (51.2 MB)
    int*   srcs    = (int*)(H + (size_t)n * HID);  // e ints
    int*   cnt     = srcs + e;                     // n
    int*   row_ptr = cnt + n;                      // n+1
    int*   cursor  = row_ptr + (n + 1);            // n
    float* dinv    = (float*)(cursor + n);         // n
    float* y       = dinv + n;                     // n
    (void)ws_size; (void)n_in; (void)out_size;

    const int TB = 256;
    init_cnt_kernel<<<(n + TB - 1) / TB, TB, 0, stream>>>(cnt, n);
    count_kernel<<<(e + TB - 1) / TB, TB, 0, stream>>>(dst, cnt, e);
    dinv_kernel<<<(n + TB - 1) / TB, TB, 0, stream>>>(cnt, dinv, n);
    scan_kernel<<<1, 1024, 0, stream>>>(cnt, row_ptr, cursor, n);
    scatter_kernel<<<(e + TB - 1) / TB, TB, 0, stream>>>(src, dst, cursor, srcs, e);

    gemm1_kernel<<<n / 16, TB, 0, stream>>>(X, W1, H);   // n % 16 == 0

    // one wave per node -> n*32 threads
    int agg_blocks = (n * 32 + TB - 1) / TB;
    agg1_dot_kernel<<<agg_blocks, TB, 0, stream>>>(H, dinv, row_ptr, srcs, b1, W2, y, n);
    agg2_kernel<<<agg_blocks, TB, 0, stream>>>(y, dinv, row_ptr, srcs, b2, out, n);
}